// GATv2Aggregator_87454124081154
// MI455X (gfx1250) — compile-verified
//
#include <hip/hip_runtime.h>
#include <hip/hip_bf16.h>

typedef __attribute__((ext_vector_type(2))) float v2f;
typedef __attribute__((ext_vector_type(8))) float v8f;

#define DIM 128
#define NEG_SLOPE 0.2f

__device__ __forceinline__ float lrelu(float x) { return x > 0.f ? x : NEG_SLOPE * x; }
__device__ __forceinline__ float elu(float x)   { return x > 0.f ? x : (__expf(x) - 1.f); }

// Monotone float<->uint encoding so atomicMax(u32) implements float max.
__device__ __forceinline__ unsigned enc_f(float f) {
    unsigned u = __float_as_uint(f);
    return (u & 0x80000000u) ? ~u : (u | 0x80000000u);
}
__device__ __forceinline__ float dec_f(unsigned u) {
    return (u & 0x80000000u) ? __uint_as_float(u & 0x7FFFFFFFu) : __uint_as_float(~u);
}

// -------------------------------------------------------------------------
// C[N x 128] = A[N x 128] * W[128 x 128]   (row-major, fp32, WMMA f32 16x16x4)
// Block = 256 threads = 8 waves. Block b -> rows [16b,16b+16); wave w -> cols
// [16w, 16w+16). N must be a multiple of 16 (50000 = 3125*16).
// -------------------------------------------------------------------------
__global__ void __launch_bounds__(256) gemm128_wmma_f32(
    const float* __restrict__ A, const float* __restrict__ W,
    float* __restrict__ C) {
    const int wave = threadIdx.x >> 5;
    const int lane = threadIdx.x & 31;
    const int half = lane >> 4;      // 0: K=k..k+1 | 1: K=k+2..k+3 (A); rows+8 on store
    const int l15  = lane & 15;
    const int row0 = blockIdx.x * 16;
    const int col0 = wave * 16;

    const float* __restrict__ Arow = A + (size_t)(row0 + l15) * DIM;

    v8f acc = {};
#pragma unroll
    for (int k = 0; k < DIM; k += 4) {
        const int ka = k + half * 2;
        v2f a, b;
        a.x = Arow[ka];
        a.y = Arow[ka + 1];
        b.x = W[(size_t)ka * DIM + col0 + l15];
        b.y = W[(size_t)(ka + 1) * DIM + col0 + l15];
        acc = __builtin_amdgcn_wmma_f32_16x16x4_f32(
            /*neg_a=*/false, a, /*neg_b=*/false, b,
            /*c_mod=*/(short)0, acc, /*reuse_a=*/false, /*reuse_b=*/false);
    }
    // C/D layout: VGPR j -> row row0 + j + half*8, col col0 + l15
#pragma unroll
    for (int j = 0; j < 8; ++j)
        C[(size_t)(row0 + j + half * 8) * DIM + col0 + l15] = acc[j];
}

// -------------------------------------------------------------------------
// Zero-fill helper (amax encoding of -inf is monotone-greater than 0, and all
// real encodings are > 0, so zero works as -inf baseline too).
// -------------------------------------------------------------------------
__global__ void zero_fill(float* __restrict__ p, int n) {
    int i = blockIdx.x * blockDim.x + threadIdx.x;
    if (i < n) p[i] = 0.f;
}

// -------------------------------------------------------------------------
// Pass 1: alpha_eh = sum_c lrelu(xl[s]+xr[d]) * att ; atomic-max per (dst,h).
// One wave per edge; lane covers 4 channels (float4). For HEADS==2 the head
// boundary is exactly the half-wave boundary.
// -------------------------------------------------------------------------
template <int HEADS>
__global__ void __launch_bounds__(256) edge_alpha(
    const int* __restrict__ srcA, const int* __restrict__ dstA, int E, int ET,
    const float* __restrict__ xl, const float* __restrict__ xr,
    const float* __restrict__ att, float* __restrict__ alpha,
    unsigned* __restrict__ amax) {
    const int e = blockIdx.x * (blockDim.x >> 5) + (threadIdx.x >> 5);
    if (e >= ET) return;
    const int lane = threadIdx.x & 31;
    int s, d;
    if (e < E) { s = srcA[e]; d = dstA[e]; } else { s = d = e - E; }

    const float4 a = ((const float4*)(xl + (size_t)s * DIM))[lane];
    const float4 b = ((const float4*)(xr + (size_t)d * DIM))[lane];
    const float4 t = ((const float4*)att)[lane];

    float p = lrelu(a.x + b.x) * t.x + lrelu(a.y + b.y) * t.y +
              lrelu(a.z + b.z) * t.z + lrelu(a.w + b.w) * t.w;

    const int span = (HEADS == 2) ? 16 : 32;   // lanes per head
#pragma unroll
    for (int m = 1; m < span; m <<= 1) p += __shfl_xor(p, m, 32);

    if ((lane & (span - 1)) == 0) {
        const int h = lane / span;
        alpha[(size_t)e * HEADS + h] = p;
        atomicMax(&amax[(size_t)d * HEADS + h], enc_f(p));
    }
}

// -------------------------------------------------------------------------
// Pass 2: w = exp(alpha - amax[dst]); atomicAdd denom; store w over alpha.
// One thread per (edge, head).
// -------------------------------------------------------------------------
template <int HEADS>
__global__ void __launch_bounds__(256) edge_exp(
    const int* __restrict__ dstA, int E, int ET,
    float* __restrict__ alpha, const unsigned* __restrict__ amax,
    float* __restrict__ denom) {
    const int i = blockIdx.x * blockDim.x + threadIdx.x;
    if (i >= ET * HEADS) return;
    const int e = i / HEADS;
    const int h = i - e * HEADS;
    const int d = (e < E) ? dstA[e] : (e - E);
    const float m = dec_f(amax[(size_t)d * HEADS + h]);
    const float w = __expf(alpha[i] - m);
    alpha[i] = w;
    atomicAdd(&denom[(size_t)d * HEADS + h], w);
}

// -------------------------------------------------------------------------
// Pass 3: out[dst] += (w / denom[dst]) * xl[src]. One wave per edge.
// -------------------------------------------------------------------------
template <int HEADS>
__global__ void __launch_bounds__(256) edge_aggregate(
    const int* __restrict__ srcA, const int* __restrict__ dstA, int E, int ET,
    const float* __restrict__ xl, const float* __restrict__ alpha,
    const float* __restrict__ denom, float* __restrict__ out) {
    const int e = blockIdx.x * (blockDim.x >> 5) + (threadIdx.x >> 5);
    if (e >= ET) return;
    const int lane = threadIdx.x & 31;
    int s, d;
    if (e < E) { s = srcA[e]; d = dstA[e]; } else { s = d = e - E; }

    const int span = (HEADS == 2) ? 16 : 32;
    const int h = lane / span;
    const float coeff = alpha[(size_t)e * HEADS + h] /
                        (denom[(size_t)d * HEADS + h] + 1e-16f);

    const float4 v = ((const float4*)(xl + (size_t)s * DIM))[lane];
    float* o = out + (size_t)d * DIM + lane * 4;
    atomicAdd(o + 0, coeff * v.x);
    atomicAdd(o + 1, coeff * v.y);
    atomicAdd(o + 2, coeff * v.z);
    atomicAdd(o + 3, coeff * v.w);
}

// -------------------------------------------------------------------------
// Epilogue: x = elu(x + bias[c]) in place.
// -------------------------------------------------------------------------
__global__ void epilogue_elu(float* __restrict__ p, const float* __restrict__ bias,
                             int n) {
    int i = blockIdx.x * blockDim.x + threadIdx.x;
    if (i < n) {
        const float v = p[i] + bias[i & (DIM - 1)];
        p[i] = elu(v);
    }
}

extern "C" void kernel_launch(void* const* d_in, const int* in_sizes, int n_in,
                              void* d_out, int out_size, void* d_ws, size_t ws_size,
                              hipStream_t stream) {
    const float* x    = (const float*)d_in[0];
    const int*   ei   = (const int*)d_in[1];
    const float* Wl1  = (const float*)d_in[2];
    const float* Wr1  = (const float*)d_in[3];
    const float* att1 = (const float*)d_in[4];
    const float* b1   = (const float*)d_in[5];
    const float* Wl2  = (const float*)d_in[6];
    const float* Wr2  = (const float*)d_in[7];
    const float* att2 = (const float*)d_in[8];
    const float* b2   = (const float*)d_in[9];

    const int N  = in_sizes[0] / DIM;      // 50000
    const int E  = in_sizes[1] / 2;        // 800000
    const int ET = E + N;                  // + self loops

    const int* srcA = ei;
    const int* dstA = ei + E;

    // Workspace layout (fp32 words)
    float*    xl    = (float*)d_ws;
    float*    xr    = xl + (size_t)N * DIM;
    float*    h1    = xr + (size_t)N * DIM;          // agg1 then hidden activations
    float*    alpha = h1 + (size_t)N * DIM;          // ET * 2 floats
    unsigned* amax  = (unsigned*)(alpha + (size_t)ET * 2);  // N * 2
    float*    denom = (float*)(amax + (size_t)N * 2);       // N * 2
    float*    out   = (float*)d_out;

    const int nNodeElems = N * DIM;
    const dim3 B(256);
    const int gNode  = (nNodeElems + 255) / 256;
    const int gStat  = (N * 4 + 255) / 256;          // amax + denom (contiguous)
    const int gGemm  = N / 16;                       // 3125, exact
    const int gWaveE = (ET + 7) / 8;                 // 8 waves / block

    // ---------------- Layer 1 (H=2, C=64, concat) ----------------
    zero_fill<<<gNode, B, 0, stream>>>(h1, nNodeElems);
    zero_fill<<<gStat, B, 0, stream>>>((float*)amax, N * 4);
    gemm128_wmma_f32<<<gGemm, B, 0, stream>>>(x, Wl1, xl);
    gemm128_wmma_f32<<<gGemm, B, 0, stream>>>(x, Wr1, xr);
    edge_alpha<2><<<gWaveE, B, 0, stream>>>(srcA, dstA, E, ET, xl, xr, att1, alpha, amax);
    edge_exp<2><<<(ET * 2 + 255) / 256, B, 0, stream>>>(dstA, E, ET, alpha, amax, denom);
    edge_aggregate<2><<<gWaveE, B, 0, stream>>>(srcA, dstA, E, ET, xl, alpha, denom, h1);
    epilogue_elu<<<gNode, B, 0, stream>>>(h1, b1, nNodeElems);

    // ---------------- Layer 2 (H=1, C=128, mean over 1 head) -----
    zero_fill<<<gNode, B, 0, stream>>>(out, nNodeElems);
    zero_fill<<<gStat, B, 0, stream>>>((float*)amax, N * 4);
    gemm128_wmma_f32<<<gGemm, B, 0, stream>>>(h1, Wl2, xl);
    gemm128_wmma_f32<<<gGemm, B, 0, stream>>>(h1, Wr2, xr);
    edge_alpha<1><<<gWaveE, B, 0, stream>>>(srcA, dstA, E, ET, xl, xr, att2, alpha, amax);
    edge_exp<1><<<(ET + 255) / 256, B, 0, stream>>>(dstA, E, ET, alpha, amax, denom);
    edge_aggregate<1><<<gWaveE, B, 0, stream>>>(srcA, dstA, E, ET, xl, alpha, denom, out);
    epilogue_elu<<<gNode, B, 0, stream>>>(out, b2, nNodeElems);
}